// CausalLoss_18863496364161
// MI455X (gfx1250) — compile-verified
//
#include <hip/hip_runtime.h>
#include <hip/hip_bf16.h>

// loss = sum_c sum_d [ S2[c,d] - S1[c,d]^2 / sc_c ] / sc_c
//   S1 = per-class feature sums, S2 = per-class feature sums of squares,
//   sc_c = max(count_c, 1)
// Segmented sums computed as one-hot matmuls via V_WMMA_F32_16X16X4_F32.

typedef __attribute__((ext_vector_type(2))) float v2f;
typedef __attribute__((ext_vector_type(8))) float v8f;

#define NN 65536
#define DD 1024
#define NCLASS 10
#define C16 16
#define COLTILES (DD / 16)      // 64
#define NCHUNKS 32              // sample chunks
#define CHUNK (NN / NCHUNKS)    // 2048 samples per wave
#define WAVES_PER_BLOCK 8
#define P1_BLOCKS ((COLTILES * NCHUNKS) / WAVES_PER_BLOCK)  // 256

// ws layout (floats): S1[16*1024] | S2[16*1024] | counts[16]
#define WS_S1 0
#define WS_S2 (C16 * DD)
#define WS_CNT (2 * C16 * DD)
#define WS_FLOATS (2 * C16 * DD + 16)

__global__ __launch_bounds__(256) void zero_ws_kernel(float* ws) {
    int i = blockIdx.x * blockDim.x + threadIdx.x;
    if (i < WS_FLOATS) ws[i] = 0.0f;
}

__global__ __launch_bounds__(256) void count_kernel(const int* __restrict__ labels,
                                                    float* __restrict__ counts) {
    __shared__ unsigned int bins[16];
    if (threadIdx.x < 16) bins[threadIdx.x] = 0u;
    __syncthreads();
    for (int i = blockIdx.x * blockDim.x + threadIdx.x; i < NN;
         i += gridDim.x * blockDim.x) {
        atomicAdd(&bins[labels[i] & 15], 1u);
    }
    __syncthreads();
    if (threadIdx.x < 16) atomicAdd(&counts[threadIdx.x], (float)bins[threadIdx.x]);
}

// Each wave: one 16-column tile, one 2048-sample chunk.
// A (16x4 one-hot): lane<16 holds K=0,1 for class-row M=lane; lane>=16 holds K=2,3 for M=lane-16.
// B (4x16):         lane<16 holds K=0,1 for col N=lane;      lane>=16 holds K=2,3 for N=lane-16.
// C (16x16):        VGPR r: lanes 0-15 -> M=r, N=lane; lanes 16-31 -> M=r+8, N=lane-16.
__global__ __launch_bounds__(256) void pass1_wmma(const float* __restrict__ x,
                                                  const int* __restrict__ labels,
                                                  float* __restrict__ S1,
                                                  float* __restrict__ S2) {
    const int lane = threadIdx.x & 31;
    const int wave = blockIdx.x * WAVES_PER_BLOCK + (threadIdx.x >> 5);
    const int coltile = wave & (COLTILES - 1);
    const int chunk = wave >> 6;
    const int d0 = coltile * 16;
    const int s_base = chunk * CHUNK;
    const int n = lane & 15;            // class row for A, feature col for B
    const int kk = (lane < 16) ? 0 : 2; // K pair handled by this half-wave

    v8f csum = {0.f, 0.f, 0.f, 0.f, 0.f, 0.f, 0.f, 0.f};
    v8f csq  = {0.f, 0.f, 0.f, 0.f, 0.f, 0.f, 0.f, 0.f};

#pragma unroll 4
    for (int it = 0; it < CHUNK; it += 4) {
        const int s = s_base + it;
        const int r0 = s + kk;
        const int l0 = labels[r0];
        const int l1 = labels[r0 + 1];
        const float b0 = x[(long)r0 * DD + d0 + n];
        const float b1 = x[(long)(r0 + 1) * DD + d0 + n];
        v2f A, B, B2;
        A.x = (l0 == n) ? 1.0f : 0.0f;
        A.y = (l1 == n) ? 1.0f : 0.0f;
        B.x = b0;       B.y = b1;
        B2.x = b0 * b0; B2.y = b1 * b1;
        csum = __builtin_amdgcn_wmma_f32_16x16x4_f32(false, A, false, B,
                                                     (short)0, csum, false, false);
        csq  = __builtin_amdgcn_wmma_f32_16x16x4_f32(false, A, false, B2,
                                                     (short)0, csq, false, false);
    }

    const int mbase = (lane < 16) ? 0 : 8;
    const int col = d0 + n;
#pragma unroll
    for (int r = 0; r < 8; ++r) {
        const int m = mbase + r;
        if (m < NCLASS) {
            atomicAdd(&S1[m * DD + col], csum[r]);
            atomicAdd(&S2[m * DD + col], csq[r]);
        }
    }
}

__global__ __launch_bounds__(256) void pass2_kernel(const float* __restrict__ S1,
                                                    const float* __restrict__ S2,
                                                    const float* __restrict__ counts,
                                                    float* __restrict__ out) {
    __shared__ float red[256];
    float acc = 0.0f;
    for (int idx = threadIdx.x; idx < NCLASS * DD; idx += 256) {
        const int c = idx >> 10;
        const int d = idx & (DD - 1);
        const float sc = fmaxf(counts[c], 1.0f);
        const float s1 = S1[c * DD + d];
        const float s2 = S2[c * DD + d];
        acc += (s2 - s1 * s1 / sc) / sc;
    }
    red[threadIdx.x] = acc;
    __syncthreads();
    for (int off = 128; off > 0; off >>= 1) {
        if (threadIdx.x < off) red[threadIdx.x] += red[threadIdx.x + off];
        __syncthreads();
    }
    if (threadIdx.x == 0) out[0] = red[0];
}

extern "C" void kernel_launch(void* const* d_in, const int* in_sizes, int n_in,
                              void* d_out, int out_size, void* d_ws, size_t ws_size,
                              hipStream_t stream) {
    const float* x = (const float*)d_in[0];
    const int* labels = (const int*)d_in[1];
    float* ws = (float*)d_ws;
    float* S1 = ws + WS_S1;
    float* S2 = ws + WS_S2;
    float* counts = ws + WS_CNT;
    float* out = (float*)d_out;

    zero_ws_kernel<<<(WS_FLOATS + 255) / 256, 256, 0, stream>>>(ws);
    count_kernel<<<64, 256, 0, stream>>>(labels, counts);
    pass1_wmma<<<P1_BLOCKS, 256, 0, stream>>>(x, labels, S1, S2);
    pass2_kernel<<<1, 256, 0, stream>>>(S1, S2, counts, out);
}